// LightGCNModel_8538394984657
// MI455X (gfx1250) — compile-verified
//
#include <hip/hip_runtime.h>
#include <hip/hip_bf16.h>

typedef __attribute__((ext_vector_type(2))) float v2f;
typedef __attribute__((ext_vector_type(8))) float v8f;

#define N_USERS   100000
#define N_ITEMS   50000
#define N_NODES   150000
#define EMB_DIM   128
#define NNZ_CT    2400000
#define ALPHA     0.25f

// d_out float offsets
#define OFF_USER    0u
#define OFF_ITEM    12800000u   // N_USERS*128
#define OFF_MASKED  19200000u   // + N_ITEMS*128
#define OFF_PRED    32000000u
#define OFF_MASK    44800000u
// scratch regions inside d_out (only valid before final outputs are written)
#define OFF_BUF_A   19200000u   // 19.2M floats
#define OFF_BUF_B   38400000u   // 19.2M floats

#define B_PITCH 136   // LDS pitch for B: 2*136 % 64 == 16 -> conflict-free half-wave reads
#define A_PITCH 132   // LDS pitch for A: conflict-free WMMA A-layout reads

// ---- CDNA5 async global->LDS copy (ASYNCcnt-tracked, no VGPR round trip) ----
__device__ __forceinline__ void async_b128_to_lds(const void* gaddr, unsigned lds_byte_addr) {
    asm volatile("global_load_async_to_lds_b128 %0, %1, off"
                 :: "v"(lds_byte_addr), "v"(gaddr)
                 : "memory");
}
__device__ __forceinline__ void wait_async0() {
    asm volatile("s_wait_asynccnt 0x0" ::: "memory");
}

// ---------------- init: cur = [user;item], combined = alpha*cur ----------------
__global__ void lgcn_init(const float* __restrict__ ue, const float* __restrict__ ie,
                          float* __restrict__ combined, float* __restrict__ cur) {
    size_t i4 = (size_t)blockIdx.x * blockDim.x + threadIdx.x;  // float4 index
    const size_t user4 = (size_t)N_USERS * EMB_DIM / 4;
    const size_t tot4  = (size_t)N_NODES * EMB_DIM / 4;
    if (i4 >= tot4) return;
    float4 v = (i4 < user4) ? ((const float4*)ue)[i4]
                            : ((const float4*)ie)[i4 - user4];
    ((float4*)cur)[i4] = v;
    float4 c; c.x = ALPHA*v.x; c.y = ALPHA*v.y; c.z = ALPHA*v.z; c.w = ALPHA*v.w;
    ((float4*)combined)[i4] = c;
}

// ---------------- zero a buffer (float4 granularity) ----------------
__global__ void lgcn_zero(float* __restrict__ p, size_t n4) {
    size_t i4 = (size_t)blockIdx.x * blockDim.x + threadIdx.x;
    if (i4 >= n4) return;
    float4 z; z.x = 0.f; z.y = 0.f; z.z = 0.f; z.w = 0.f;
    ((float4*)p)[i4] = z;
}

// ---------------- SpMM: 32 edges per wave32, readlane-broadcast metadata ------
// Lane L preloads metadata for edge e0+L (coalesced), then the wave processes
// the 32 edges one by one: per edge, each lane handles a float4 of the 128-dim
// row (gather from cur[col], global_atomic_add_f32 scatter into next[row]).
__global__ void lgcn_spmm(const int* __restrict__ rows, const int* __restrict__ cols,
                          const float* __restrict__ vals, const float* __restrict__ x,
                          float* __restrict__ y, int nnz) {
    const int lane  = threadIdx.x & 31;
    const int waveG = (int)((blockIdx.x * blockDim.x + threadIdx.x) >> 5);
    const int e0    = waveG * 32;
    if (e0 >= nnz) return;
    int myE = e0 + lane;
    int r = 0, c = 0; float v = 0.f;
    if (myE < nnz) { r = rows[myE]; c = cols[myE]; v = vals[myE]; }
    const int cnt = (nnz - e0) < 32 ? (nnz - e0) : 32;
    for (int j = 0; j < cnt; ++j) {
        int   rj = __shfl(r, j);
        int   cj = __shfl(c, j);
        float vj = __shfl(v, j);
        float4 xv = ((const float4*)(x + (size_t)cj * EMB_DIM))[lane];
        float* yp = y + (size_t)rj * EMB_DIM + lane * 4;
        unsafeAtomicAdd(yp + 0, vj * xv.x);
        unsafeAtomicAdd(yp + 1, vj * xv.y);
        unsafeAtomicAdd(yp + 2, vj * xv.z);
        unsafeAtomicAdd(yp + 3, vj * xv.w);
    }
}

// ---------------- combined += alpha * next ----------------
__global__ void lgcn_axpy(float* __restrict__ combined, const float* __restrict__ nxt, size_t n4) {
    size_t i4 = (size_t)blockIdx.x * blockDim.x + threadIdx.x;
    if (i4 >= n4) return;
    float4 c = ((const float4*)combined)[i4];
    float4 v = ((const float4*)nxt)[i4];
    c.x += ALPHA*v.x; c.y += ALPHA*v.y; c.z += ALPHA*v.z; c.w += ALPHA*v.w;
    ((float4*)combined)[i4] = c;
}

// ---------------- masked = user_final * mask ; mask_out = mask ----------------
__global__ void lgcn_mask(const float* __restrict__ userf, const float* __restrict__ mask,
                          float* __restrict__ masked, float* __restrict__ mask_out, size_t n4) {
    size_t i4 = (size_t)blockIdx.x * blockDim.x + threadIdx.x;
    if (i4 >= n4) return;
    float4 u = ((const float4*)userf)[i4];
    float4 m = ((const float4*)mask)[i4];
    float4 o; o.x = u.x*m.x; o.y = u.y*m.y; o.z = u.z*m.z; o.w = u.w*m.w;
    ((float4*)masked)[i4]   = o;
    ((float4*)mask_out)[i4] = m;
}

// ---------------- GEMM: relu(masked @ W + b) via V_WMMA_F32_16X16X4_F32 --------
// Block: 256 threads = 8 waves. Block handles a 16-row M tile; wave w handles
// N tile w (16 cols). A tile (16x128) and the full B matrix (128x128) are
// staged in LDS with async global->LDS b128 copies; inner loop is DS+WMMA only.
__global__ void lgcn_gemm(const float* __restrict__ A, const float* __restrict__ B,
                          const float* __restrict__ bias, float* __restrict__ out) {
    __shared__ float As[16 * A_PITCH];        //  8,448 B
    __shared__ float Bs[EMB_DIM * B_PITCH];   // 69,632 B
    const int tid   = threadIdx.x;
    const int wave  = tid >> 5;      // 0..7 -> N tile
    const int lane  = tid & 31;
    const int mTile = blockIdx.x;

    // Async stage A tile: 16 rows x 128 = 512 float4, coalesced
    const float4* A4 = (const float4*)(A + (size_t)mTile * 16 * EMB_DIM);
    #pragma unroll
    for (int i = tid; i < 512; i += 256) {
        int row = i >> 5;            // 32 float4 per row
        int c4  = i & 31;
        unsigned dst = (unsigned)(size_t)&As[row * A_PITCH + c4 * 4];
        async_b128_to_lds(A4 + i, dst);
    }
    // Async stage B: 128x128 = 4096 float4, coalesced
    const float4* B4 = (const float4*)B;
    #pragma unroll
    for (int i = tid; i < 4096; i += 256) {
        int k  = i >> 5;
        int c4 = i & 31;
        unsigned dst = (unsigned)(size_t)&Bs[k * B_PITCH + c4 * 4];
        async_b128_to_lds(B4 + i, dst);
    }
    wait_async0();        // own async copies complete
    __syncthreads();      // everyone's copies visible

    const int m    = lane & 15;      // A row / D col within tile
    const int half = lane >> 4;      // K sub-offset selector
    const int n    = lane & 15;      // B/D column within tile
    const float* ArowP = &As[m * A_PITCH + half * 2];
    const float* BcolP = &Bs[half * 2 * B_PITCH + wave * 16 + n];

    v8f acc = {};
    #pragma unroll
    for (int k0 = 0; k0 < EMB_DIM; k0 += 4) {
        // A layout (ISA 7.12.2, 32-bit A 16x4): lanes 0-15 K={0,1}; lanes 16-31 K={2,3}
        float2 av = *(const float2*)&ArowP[k0];
        v2f a; a.x = av.x; a.y = av.y;
        // B layout mirrors A striping: VGPR0 -> K=0/2, VGPR1 -> K=1/3, N = lane%16
        v2f b;
        b.x = BcolP[(size_t)k0 * B_PITCH];
        b.y = BcolP[(size_t)(k0 + 1) * B_PITCH];
        acc = __builtin_amdgcn_wmma_f32_16x16x4_f32(
                  false, a, false, b, (short)0, acc, false, false);
    }

    // Epilogue: D VGPR r -> row r (lanes 0-15) / r+8 (lanes 16-31); col = lane%16
    float bv = bias[wave * 16 + n];
    float* outp = out + (size_t)mTile * 16 * EMB_DIM + wave * 16 + n;
    #pragma unroll
    for (int r = 0; r < 8; ++r) {
        int mr = r + half * 8;
        float v = acc[r] + bv;
        v = v > 0.f ? v : 0.f;
        outp[(size_t)mr * EMB_DIM] = v;
    }
}

extern "C" void kernel_launch(void* const* d_in, const int* in_sizes, int n_in,
                              void* d_out, int out_size, void* d_ws, size_t ws_size,
                              hipStream_t stream) {
    const float* user_emb = (const float*)d_in[0];
    const float* item_emb = (const float*)d_in[1];
    const int*   adj_rows = (const int*)d_in[2];
    const int*   adj_cols = (const int*)d_in[3];
    const float* adj_vals = (const float*)d_in[4];
    const float* mask     = (const float*)d_in[5];
    const float* attr_w   = (const float*)d_in[6];
    const float* attr_b   = (const float*)d_in[7];
    (void)in_sizes; (void)n_in; (void)out_size; (void)d_ws; (void)ws_size;

    float* out      = (float*)d_out;
    float* combined = out + OFF_USER;     // 19.2M floats (user_final || item_final)
    float* bufA     = out + OFF_BUF_A;    // scratch: later masked+pred[0:6.4M]
    float* bufB     = out + OFF_BUF_B;    // scratch: later pred[6.4M:]+mask_out

    const size_t node4 = (size_t)N_NODES * EMB_DIM / 4;   // 4.8M float4
    const size_t user4 = (size_t)N_USERS * EMB_DIM / 4;   // 3.2M float4
    const int nnz = NNZ_CT;

    dim3 blk(256);
    dim3 gNode((unsigned)((node4 + 255) / 256));
    dim3 gUser((unsigned)((user4 + 255) / 256));
    // one wave32 per 32 edges
    const size_t spmmThreads = ((size_t)nnz + 31) / 32 * 32;
    dim3 gSpmm((unsigned)((spmmThreads + 255) / 256));
    dim3 gGemm(N_USERS / 16);

    // combined = alpha*emb ; cur = emb
    lgcn_init<<<gNode, blk, 0, stream>>>(user_emb, item_emb, combined, bufA);

    float* cur = bufA;
    float* nxt = bufB;
    for (int layer = 0; layer < 3; ++layer) {
        lgcn_zero<<<gNode, blk, 0, stream>>>(nxt, node4);
        lgcn_spmm<<<gSpmm, blk, 0, stream>>>(adj_rows, adj_cols, adj_vals, cur, nxt, nnz);
        lgcn_axpy<<<gNode, blk, 0, stream>>>(combined, nxt, node4);
        float* t = cur; cur = nxt; nxt = t;
    }

    // masked_user (overwrites scratch A) and mask_out (overwrites tail of scratch B)
    lgcn_mask<<<gUser, blk, 0, stream>>>(out + OFF_USER, mask,
                                         out + OFF_MASKED, out + OFF_MASK, user4);
    // predicted = relu(masked @ W + b) — fp32 WMMA
    lgcn_gemm<<<gGemm, blk, 0, stream>>>(out + OFF_MASKED, attr_w, attr_b, out + OFF_PRED);
}